// Bottomk_CrossEntropyLoss_19619410608218
// MI455X (gfx1250) — compile-verified
//
#include <hip/hip_runtime.h>
#include <hip/hip_bf16.h>
#include <stdint.h>

// Problem constants (match reference setup_inputs / K=0.5)
#define NROWS       8192
#define VDIM        32000
#define TPB         256                       // 8 waves (wave32)
#define TILE_FLOATS 1024                      // 256 lanes * 4 floats (B128 per lane)
#define TILE_BYTES  (TILE_FLOATS * 4)         // 4096
#define NTILES      31                        // 31 * 1024 = 31744
#define REM_BASE    (NTILES * TILE_FLOATS)    // 31744
#define REM         (VDIM - REM_BASE)         // 256 == TPB
#define DEPTH       4                         // async pipeline depth (4 LDS buffers)
#define SEL_T       1024
#define MKEEP       (NROWS / 2)               // int(N * 0.5) = 4096

static_assert(REM == TPB, "tail sized to one element per thread");
static_assert((DEPTH & (DEPTH - 1)) == 0, "DEPTH power of two");

// ---------------------------------------------------------------------------
// Kernel 1: per-row logsumexp - picked  ->  ce[row]
// Streaming via CDNA5 async global->LDS loads, 4-deep pipeline:
// 3 tiles in flight per wave while one tile is consumed (in-order ASYNCcnt).
// ---------------------------------------------------------------------------
__global__ __launch_bounds__(TPB) void ce_rows_kernel(const float* __restrict__ logits,
                                                      const int*   __restrict__ labels,
                                                      float*       __restrict__ ce)
{
    __shared__ float buf[DEPTH][TILE_FLOATS];   // 16 KB
    __shared__ float red_m[TPB / 32];
    __shared__ float red_s[TPB / 32];

    const int row = blockIdx.x;
    const int tid = threadIdx.x;

    const uint64_t rowbase  = (uint64_t)(uintptr_t)(logits + (size_t)row * VDIM);
    // addrspace(3) offsets: low 32 bits of the generic pointer are the LDS offset
    const uint32_t lds_base = (uint32_t)(uintptr_t)(&buf[0][tid * 4]);

    // Prime the pipeline: tiles 0..DEPTH-1 into buffers 0..DEPTH-1.
#pragma unroll
    for (int p = 0; p < DEPTH; ++p) {
        uint64_t a = rowbase + (uint64_t)(p * TILE_BYTES + tid * 16);
        uint32_t l = lds_base + (uint32_t)(p * TILE_BYTES);
        asm volatile("global_load_async_to_lds_b128 %0, %1, off"
                     :: "v"(l), "v"(a) : "memory");
    }

    float m = -__builtin_inff();
    float s = 0.0f;

    for (int t = 0; t < NTILES; ++t) {
        // Per-wave async loads complete in order; tile t is done once at most
        // (tiles issued beyond t) remain outstanding.
        if (t <= NTILES - DEPTH) {
            asm volatile("s_wait_asynccnt 0x3" ::: "memory");
        } else if (t == NTILES - 3) {
            asm volatile("s_wait_asynccnt 0x2" ::: "memory");
        } else if (t == NTILES - 2) {
            asm volatile("s_wait_asynccnt 0x1" ::: "memory");
        } else {
            asm volatile("s_wait_asynccnt 0x0" ::: "memory");
        }

        const float4 x = *reinterpret_cast<const float4*>(&buf[t & (DEPTH - 1)][tid * 4]);
        const float tmax = fmaxf(fmaxf(x.x, x.y), fmaxf(x.z, x.w));
        if (tmax > m) {              // rare after warm-up: one rescale per tile at most
            s *= __expf(m - tmax);   // m == -inf on first tile -> s stays 0 (exp(-inf)=0)
            m = tmax;
        }
        s += __expf(x.x - m) + __expf(x.y - m) + __expf(x.z - m) + __expf(x.w - m);

        // Refill this buffer with tile t+DEPTH once our LDS reads have drained.
        if (t + DEPTH < NTILES) {
            uint64_t a = rowbase + (uint64_t)((t + DEPTH) * TILE_BYTES + tid * 16);
            uint32_t l = lds_base + (uint32_t)((t & (DEPTH - 1)) * TILE_BYTES);
            asm volatile("s_wait_dscnt 0x0\n\t"
                         "global_load_async_to_lds_b128 %0, %1, off"
                         :: "v"(l), "v"(a) : "memory");
        }
    }

    // Tail: 256 floats, one per thread, straight from global.
    {
        const float x = logits[(size_t)row * VDIM + REM_BASE + tid];
        if (x > m) { s *= __expf(m - x); m = x; }
        s += __expf(x - m);
    }

    // Wave32 reduction of (m, s) pairs — fixed order, deterministic.
#pragma unroll
    for (int off = 16; off > 0; off >>= 1) {
        const float m2 = __shfl_xor(m, off, 32);
        const float s2 = __shfl_xor(s, off, 32);
        const float M  = fmaxf(m, m2);
        s = s * __expf(m - M) + s2 * __expf(m2 - M);
        m = M;
    }
    const int wave = tid >> 5;
    if ((tid & 31) == 0) { red_m[wave] = m; red_s[wave] = s; }
    __syncthreads();

    if (tid == 0) {
        float M = red_m[0], S = red_s[0];
#pragma unroll
        for (int w = 1; w < TPB / 32; ++w) {
            const float m2 = red_m[w], s2 = red_s[w];
            const float Mn = fmaxf(M, m2);
            S = S * __expf(M - Mn) + s2 * __expf(m2 - Mn);
            M = Mn;
        }
        const float picked = logits[(size_t)row * VDIM + (size_t)labels[row]];
        ce[row] = M + __logf(S) - picked;
    }
}

// ---------------------------------------------------------------------------
// Kernel 2: exact bottom-MKEEP mean via 32-bit radix select in LDS.
// ---------------------------------------------------------------------------
__device__ __forceinline__ unsigned f2ord(float f) {
    unsigned u = __float_as_uint(f);
    return u ^ (unsigned)(((int)u >> 31) | 0x80000000);
}
__device__ __forceinline__ float ord2f(unsigned k) {
    unsigned u = (k & 0x80000000u) ? (k ^ 0x80000000u) : ~k;
    return __uint_as_float(u);
}

__global__ __launch_bounds__(SEL_T) void bottomk_mean_kernel(const float* __restrict__ ce,
                                                             float*       __restrict__ outv)
{
    __shared__ unsigned keys[NROWS];   // 32 KB of the 320 KB WGP LDS
    __shared__ int      cnt;
    __shared__ float    ps[SEL_T];
    __shared__ int      pc[SEL_T];

    const int tid = threadIdx.x;
    for (int i = tid; i < NROWS; i += SEL_T) keys[i] = f2ord(ce[i]);
    __syncthreads();

    // Find the exact bit pattern of the MKEEP-th smallest key (rank select, MSB->LSB).
    unsigned prefix = 0, hmask = 0;
    int remaining = MKEEP;
#pragma unroll 1
    for (int b = 31; b >= 0; --b) {
        const unsigned bit = 1u << b;
        if (tid == 0) cnt = 0;
        __syncthreads();
        int local = 0;
        for (int i = tid; i < NROWS; i += SEL_T) {
            const unsigned k = keys[i];
            local += (int)(((k & hmask) == prefix) && ((k & bit) == 0));
        }
        atomicAdd(&cnt, local);          // integer atomic: deterministic
        __syncthreads();
        const int c0 = cnt;
        if (c0 < remaining) { remaining -= c0; prefix |= bit; }
        hmask |= bit;
        __syncthreads();
    }
    const unsigned T = prefix;           // key of the MKEEP-th smallest value

    // Sum strictly-smaller values; pad ties with value(T). Fixed-order tree reduce.
    float lsum = 0.0f;
    int   lcnt = 0;
    for (int i = tid; i < NROWS; i += SEL_T) {
        const unsigned k = keys[i];
        if (k < T) { lsum += ord2f(k); ++lcnt; }
    }
    ps[tid] = lsum; pc[tid] = lcnt;
    __syncthreads();
    for (int off = SEL_T / 2; off > 0; off >>= 1) {
        if (tid < off) { ps[tid] += ps[tid + off]; pc[tid] += pc[tid + off]; }
        __syncthreads();
    }
    if (tid == 0) {
        const float total = ps[0] + (float)(MKEEP - pc[0]) * ord2f(T);
        outv[0] = total / (float)MKEEP;
    }
}

// ---------------------------------------------------------------------------
extern "C" void kernel_launch(void* const* d_in, const int* in_sizes, int n_in,
                              void* d_out, int out_size, void* d_ws, size_t ws_size,
                              hipStream_t stream) {
    (void)in_sizes; (void)n_in; (void)out_size; (void)ws_size;
    const float* logits = (const float*)d_in[0];
    const int*   labels = (const int*)d_in[1];
    float* cew = (float*)d_ws;     // NROWS floats of scratch
    float* res = (float*)d_out;    // single f32 result

    ce_rows_kernel<<<NROWS, TPB, 0, stream>>>(logits, labels, cew);
    bottomk_mean_kernel<<<1, SEL_T, 0, stream>>>(cew, res);
}